// DGCN_84559316123789
// MI455X (gfx1250) — compile-verified
//
#include <hip/hip_runtime.h>

typedef __attribute__((ext_vector_type(2))) float v2f;
typedef __attribute__((ext_vector_type(4))) float v4f;
typedef __attribute__((ext_vector_type(8))) float v8f;
typedef __attribute__((ext_vector_type(4))) int   v4i;

#define DG_B 8
#define DG_T 12
#define DG_N 2048
#define DG_C 64
#define TM   128
#define BK   16
#define ADJ_LDW 20   // adjT row stride: 20*dm never ≡ 0,±1,2,3 mod 64 -> conflict-free
#define Y_LDW   68   // Yt row stride: 4*dm never ≡ 1..3 mod 64 -> conflict-free halves
#define XP_LDW  160  // pair-interleaved row-pair stride (≡32 mod 64): halves split banks
#define WP_LDW  160
#define STAGE_FLOATS (TM * ADJ_LDW + (BK / 2) * XP_LDW)   // 2560 + 1280 = 3840

// CDNA5 async global->LDS path (guarded: falls back to load+ds_store pipeline).
// Probe-confirmed: builtin exists; param0 is v4i* in AS1 (global), dst in AS3.
#if defined(__AMDGCN__) && __has_builtin(__builtin_amdgcn_global_load_async_to_lds_b128)
#define DG_ASYNC 1
typedef __attribute__((address_space(1))) v4i dg_glb_v4i;
typedef __attribute__((address_space(3))) v4i dg_lds_v4i;
#else
#define DG_ASYNC 0
#endif

__device__ __forceinline__ void dg_wait_async() {
#if DG_ASYNC
#if __has_builtin(__builtin_amdgcn_s_wait_asynccnt)
  __builtin_amdgcn_s_wait_asynccnt(0);
#else
  asm volatile("s_wait_asynccnt 0" ::: "memory");
#endif
#endif
}

// ---------------------------------------------------------------------------
// Pass A: lambda MLP per time-step, then write c1[t]*W0 and c2[t]*W1 to ws.
// ---------------------------------------------------------------------------
__global__ void dgcn_lambda_weights(const float* __restrict__ st_emb,
                                    const float* __restrict__ W1,
                                    const float* __restrict__ b1,
                                    const float* __restrict__ W2,
                                    const float* __restrict__ b2,
                                    const float* __restrict__ weights,
                                    float* __restrict__ W0s,
                                    float* __restrict__ W1s) {
  const int t = blockIdx.x;
  const int tid = threadIdx.x;
  __shared__ float h[32];
  __shared__ float cc[2];
  if (tid < 32) {
    float s = b1[tid];
    for (int i = 0; i < 64; ++i) s = fmaf(st_emb[t * 64 + i], W1[i * 32 + tid], s);
    h[tid] = fmaxf(s, 0.0f);
  }
  __syncthreads();
  if (tid == 0) {
    float s = b2[0];
    for (int j = 0; j < 32; ++j) s = fmaf(h[j], W2[j], s);
    const float lam = 1.0f + fmaxf(s, 0.0f);
    cc[0] = (2.0f * lam - 2.0f) / lam;   // c1
    cc[1] = 2.0f / lam;                  // c2
  }
  __syncthreads();
  const float c1 = cc[0], c2 = cc[1];
  for (int i = tid; i < 64 * 64; i += blockDim.x) {
    W0s[t * 4096 + i] = c1 * weights[i];          // weights[0] (identity support)
    W1s[t * 4096 + i] = c2 * weights[4096 + i];   // weights[1] (adjacency support)
  }
}

// ---------------------------------------------------------------------------
// Main kernel, double-buffered stage 1:
//   Y = adj_tile @ X   (K = 2048 in 128 chunks of 16; fetch of chunk i+1
//                       overlaps WMMA of chunk i; adj tile goes global->LDS
//                       via async-to-LDS when available)
//   out = X @ (c1 W0) + Y @ (c2 W1) + bias   (two K=64 WMMA epilogue GEMMs)
// B operands pair-interleaved in LDS: elem(k,n) -> [(k>>1)*S + 2n + (k&1)]
// so each B fragment is ONE aligned ds_load_b64.
// ---------------------------------------------------------------------------
__global__ __launch_bounds__(256) void dgcn_main(
    const float* __restrict__ x, const float* __restrict__ adj,
    const float* __restrict__ W0s, const float* __restrict__ W1s,
    const float* __restrict__ bias, float* __restrict__ out) {
  // Region: two stage buffers (2*3840 floats) aliased with Yt[128][68]
  // (8704 floats, the larger), + Wt (5120). Total 13824 floats = 55.3 KB.
  __shared__ __align__(16) float smem[TM * Y_LDW + 32 * WP_LDW];
  float* const buf0 = smem;
  float* const buf1 = smem + STAGE_FLOATS;
  float* const Yt   = smem;                    // reuses stage region
  float* const Wt   = smem + TM * Y_LDW;       // pair-interleaved [32][WP_LDW]

  const int tid     = threadIdx.x;
  const int wave    = tid >> 5;
  const int lane    = tid & 31;
  const int l16     = lane & 15;
  const int khalf   = (lane < 16) ? 0 : 2;
  const int mhalf   = (lane < 16) ? 0 : 8;
  const int rowbase = wave * 16;               // each wave: 16-row band, 64 cols

  const int n0 = blockIdx.x * TM;
  const int bt = blockIdx.y;
  const int t  = bt % DG_T;

  const float* const xbt = x + (size_t)bt * DG_N * DG_C;

  // Preload Wt = c2*W1 (used first in the epilogue), pair-interleaved.
  {
    const float* const w1 = W1s + t * 4096;
    for (int i = tid * 4; i < 4096; i += 256 * 4) {
      const v4f w = *(const v4f*)&w1[i];
      const int k = i >> 6, c = i & 63;
      float* const dst = &Wt[(k >> 1) * WP_LDW + (k & 1) + 2 * c];
      dst[0] = w.x; dst[2] = w.y; dst[4] = w.z; dst[6] = w.w;
    }
  }

  // ---- Per-thread tile-copy assignments -----------------------------------
  // adj tile: 128 rows x 16 floats = 512 float4; 2 per thread.
  const int fa0 = tid, fa1 = tid + 256;
  const int ar0 = fa0 >> 2, ac0 = (fa0 & 3) << 2;
  const int ar1 = fa1 >> 2, ac1 = (fa1 & 3) << 2;
  const int ldsA0 = ar0 * ADJ_LDW + ac0;
  const int ldsA1 = ar1 * ADJ_LDW + ac1;
  const float* ag0 = adj + (size_t)(n0 + ar0) * DG_N + ac0;
  const float* ag1 = adj + (size_t)(n0 + ar1) * DG_N + ac1;
  // x tile: 16 rows x 64 floats = 256 float4; 1 per thread (interleave scatter)
  const int xr = tid >> 4, xc = (tid & 15) << 2;
  const int ldsX = (xr >> 1) * XP_LDW + (xr & 1) + 2 * xc;
  const float* xg = xbt + (size_t)xr * 64 + xc;

  v4f xv;
#if !DG_ASYNC
  v4f av0, av1;
#endif

  auto issue = [&](float* sb) {           // fetch next chunk (overlaps compute)
#if DG_ASYNC
    __builtin_amdgcn_global_load_async_to_lds_b128(
        (dg_glb_v4i*)ag0, (dg_lds_v4i*)(sb + ldsA0), 0, 0);
    __builtin_amdgcn_global_load_async_to_lds_b128(
        (dg_glb_v4i*)ag1, (dg_lds_v4i*)(sb + ldsA1), 0, 0);
#else
    av0 = *(const v4f*)ag0;
    av1 = *(const v4f*)ag1;
#endif
    xv = *(const v4f*)xg;
    ag0 += BK; ag1 += BK; xg += BK * 64;
  };
  auto finish = [&](float* sb) {          // complete staging of next chunk
#if !DG_ASYNC
    *(v4f*)(sb + ldsA0) = av0;
    *(v4f*)(sb + ldsA1) = av1;
#endif
    float* const dst = sb + TM * ADJ_LDW + ldsX;
    dst[0] = xv.x; dst[2] = xv.y; dst[4] = xv.z; dst[6] = xv.w;
    dg_wait_async();
  };

  const v8f vzero = {0, 0, 0, 0, 0, 0, 0, 0};
  v8f acc[4];
  acc[0] = vzero; acc[1] = vzero; acc[2] = vzero; acc[3] = vzero;

  auto compute = [&](const float* sb) {
    const float* const adjT = sb;
    const float* const xT   = sb + TM * ADJ_LDW;
#pragma unroll
    for (int kk = 0; kk < BK; kk += 4) {
      const int kb = kk + khalf;               // even
      const v2f a = *(const v2f*)&adjT[(rowbase + l16) * ADJ_LDW + kb];
#pragma unroll
      for (int cg = 0; cg < 4; ++cg) {
        const v2f bv = *(const v2f*)&xT[(kb >> 1) * XP_LDW + 2 * (cg * 16 + l16)];
        acc[cg] = __builtin_amdgcn_wmma_f32_16x16x4_f32(
            false, a, false, bv, (short)0, acc[cg], false, false);
      }
    }
  };

  // ---- Stage 1 pipeline: 128 chunks, 2 buffers ----------------------------
  issue(buf0);                // chunk 0
  finish(buf0);
  __syncthreads();
  for (int c = 0; c < 126; c += 2) {
    issue(buf1);              // chunk c+1
    compute(buf0);            // chunk c
    finish(buf1);
    __syncthreads();
    issue(buf0);              // chunk c+2
    compute(buf1);            // chunk c+1
    finish(buf0);
    __syncthreads();
  }
  issue(buf1);                // chunk 127
  compute(buf0);              // chunk 126
  finish(buf1);
  __syncthreads();
  compute(buf1);              // chunk 127

  // ---- Spill Y to LDS (C layout -> row-major), reusing stage region -------
  __syncthreads();
#pragma unroll
  for (int cg = 0; cg < 4; ++cg) {
#pragma unroll
    for (int j = 0; j < 8; ++j)
      Yt[(rowbase + j + mhalf) * Y_LDW + cg * 16 + l16] = acc[cg][j];
  }
  __syncthreads();

  acc[0] = vzero; acc[1] = vzero; acc[2] = vzero; acc[3] = vzero;

  // ---- Epilogue part 1: Y @ (c2*W1), K=64, A from LDS ---------------------
#pragma unroll
  for (int kk = 0; kk < 64; kk += 4) {
    const int kb = kk + khalf;
    const v2f a = *(const v2f*)&Yt[(rowbase + l16) * Y_LDW + kb];
#pragma unroll
    for (int cg = 0; cg < 4; ++cg) {
      const v2f bv = *(const v2f*)&Wt[(kb >> 1) * WP_LDW + 2 * (cg * 16 + l16)];
      acc[cg] = __builtin_amdgcn_wmma_f32_16x16x4_f32(
          false, a, false, bv, (short)0, acc[cg], false, false);
    }
  }

  // Swap Wt -> c1*W0.
  __syncthreads();
  {
    const float* const w0 = W0s + t * 4096;
    for (int i = tid * 4; i < 4096; i += 256 * 4) {
      const v4f w = *(const v4f*)&w0[i];
      const int k = i >> 6, c = i & 63;
      float* const dst = &Wt[(k >> 1) * WP_LDW + (k & 1) + 2 * c];
      dst[0] = w.x; dst[2] = w.y; dst[4] = w.z; dst[6] = w.w;
    }
  }
  __syncthreads();

  // ---- Epilogue part 2: X @ (c1*W0), K=64, A-frags straight from global ---
  const float* const xrow = xbt + (size_t)(n0 + rowbase + l16) * 64;
#pragma unroll
  for (int kk = 0; kk < 64; kk += 4) {
    const int kb = kk + khalf;
    const v2f a = *(const v2f*)&xrow[kb];
#pragma unroll
    for (int cg = 0; cg < 4; ++cg) {
      const v2f bv = *(const v2f*)&Wt[(kb >> 1) * WP_LDW + 2 * (cg * 16 + l16)];
      acc[cg] = __builtin_amdgcn_wmma_f32_16x16x4_f32(
          false, a, false, bv, (short)0, acc[cg], false, false);
    }
  }

  // ---- Bias + store (C layout: lane=col, VGPR=row) ------------------------
  float* const obt = out + (size_t)bt * DG_N * DG_C;
#pragma unroll
  for (int cg = 0; cg < 4; ++cg) {
    const int nn = cg * 16 + l16;
    const float bv = bias[nn];
#pragma unroll
    for (int j = 0; j < 8; ++j) {
      const int m = n0 + rowbase + j + mhalf;
      obt[(size_t)m * 64 + nn] = acc[cg][j] + bv;
    }
  }
}

// ---------------------------------------------------------------------------
extern "C" void kernel_launch(void* const* d_in, const int* in_sizes, int n_in,
                              void* d_out, int out_size, void* d_ws, size_t ws_size,
                              hipStream_t stream) {
  (void)in_sizes; (void)n_in; (void)out_size; (void)ws_size;
  const float* x       = (const float*)d_in[0];
  const float* adj     = (const float*)d_in[1];
  const float* st_emb  = (const float*)d_in[2];
  const float* weights = (const float*)d_in[3];
  const float* bias    = (const float*)d_in[4];
  const float* W1      = (const float*)d_in[5];
  const float* b1      = (const float*)d_in[6];
  const float* W2      = (const float*)d_in[7];
  const float* b2      = (const float*)d_in[8];
  float* out = (float*)d_out;

  float* W0s = (float*)d_ws;              // [T][64][64] c1-scaled
  float* W1s = W0s + DG_T * 64 * 64;      // [T][64][64] c2-scaled

  dgcn_lambda_weights<<<DG_T, 64, 0, stream>>>(st_emb, W1, b1, W2, b2,
                                               weights, W0s, W1s);
  dim3 grid(DG_N / TM, DG_B * DG_T);
  dgcn_main<<<grid, 256, 0, stream>>>(x, adj, W0s, W1s, bias, out);
}